// TransitionModule_5970004542260
// MI455X (gfx1250) — compile-verified
//
#include <hip/hip_runtime.h>
#include <cmath>

// ---------------------------------------------------------------------------
// TransitionModule: out[b] = tanh(W[action[b]] @ g_prev[b] + bias)
// B=128, H=1024, A=64, all f32.  HBM-bound (~256MB weights vs 268 MFLOP):
//  - samples grouped by action => each W matrix streamed from HBM once
//  - double-buffered LDS tiles, async global->LDS copies (ASYNCcnt pipeline)
//  - V_WMMA_F32_16X16X4_F32 accumulation (exact f32 math)
// ---------------------------------------------------------------------------

typedef __attribute__((ext_vector_type(2))) float v2f;
typedef __attribute__((ext_vector_type(8))) float v8f;
typedef __attribute__((ext_vector_type(4))) int   v4i;

#define B_SAMPLES 128
#define H_DIM     1024
#define A_ACT     64
#define ROWS_PB   128              // 8 waves * 16 rows per block
#define KT        64               // k-tile staged in LDS
#define COLS      16               // WMMA N dimension (samples per tile)
#define NT        (H_DIM / KT)     // 16 k-tiles
#define PAD       4                // row pad: 16B-aligned stride, bank-safe
#define LROW      (KT + PAD)       // 68 floats = 272B row stride

#if __has_builtin(__builtin_amdgcn_global_load_async_to_lds_b128)
#define HAVE_ASYNC_LDS 1
#else
#define HAVE_ASYNC_LDS 0
#endif

// builtin signature (from hipcc diagnostic): (v4i addrspace(1)*, v4i addrspace(3)*, Imm, Imm)
typedef v4i __attribute__((address_space(1))) g_v4i;
typedef v4i __attribute__((address_space(3))) l_v4i;

// 16B global -> LDS copy; async (ASYNCcnt) when available.
__device__ __forceinline__ void copy16_g2l(const float* g, float* l) {
#if HAVE_ASYNC_LDS
    __builtin_amdgcn_global_load_async_to_lds_b128((g_v4i*)g, (l_v4i*)l,
                                                   /*offset=*/0, /*cpol=*/0);
#else
    *(float4*)l = *(const float4*)g;
#endif
}

__device__ __forceinline__ void wait_async_le9() {
#if HAVE_ASYNC_LDS
#if __has_builtin(__builtin_amdgcn_s_wait_asynccnt)
    __builtin_amdgcn_s_wait_asynccnt(9);
#else
    asm volatile("s_wait_asynccnt 9" ::: "memory");
#endif
#endif
}

__device__ __forceinline__ void wait_async_0() {
#if HAVE_ASYNC_LDS
#if __has_builtin(__builtin_amdgcn_s_wait_asynccnt)
    __builtin_amdgcn_s_wait_asynccnt(0);
#else
    asm volatile("s_wait_asynccnt 0" ::: "memory");
#endif
#endif
}

// d_ws layout (ints): counts[A_ACT], lists[A_ACT][B_SAMPLES] (zero-padded)
__global__ void tm_build_lists(const int* __restrict__ idx, int* __restrict__ ws) {
    // zero-pad lists so padded sample slots hold a valid index (0)
    for (int i = threadIdx.x; i < A_ACT * B_SAMPLES; i += blockDim.x)
        ws[A_ACT + i] = 0;
    __syncthreads();
    if (threadIdx.x == 0) {
        int* counts = ws;
        int* lists  = ws + A_ACT;
        for (int a = 0; a < A_ACT; ++a) counts[a] = 0;
        for (int b = 0; b < B_SAMPLES; ++b) {      // stable => deterministic
            int a = idx[b] & (A_ACT - 1);
            lists[a * B_SAMPLES + counts[a]] = b;
            counts[a]++;
        }
    }
}

__launch_bounds__(256)
__global__ void tm_wmma_kernel(const float* __restrict__ g_prev,   // [B][H]
                               const float* __restrict__ weights,  // [A][H][H]
                               const float* __restrict__ bias,     // [H]
                               const int*   __restrict__ ws,       // counts+lists
                               float*       __restrict__ out)      // [B][H]
{
    __shared__ float Ws[2][ROWS_PB][LROW];   // 2 x 34816B
    __shared__ float Gs[2][COLS][LROW];      // 2 x  4352B

    const int a  = blockIdx.x;
    const int n  = ws[a];
    if (n == 0) return;                      // uniform across block
    const int r0 = blockIdx.y * ROWS_PB;
    const int* list = ws + A_ACT + a * B_SAMPLES;

    const int tid   = threadIdx.x;
    const int lane  = tid & 31;
    const int wave  = tid >> 5;
    const int m0    = wave * 16;             // this wave's rows within tile
    const int lrow  = lane & 15;             // M (A-frag) / N (B-frag)
    const int khalf = (lane >> 4) << 1;      // lanes 0-15 -> K+0,1 ; 16-31 -> K+2,3

    const float* Wa = weights + (size_t)a * H_DIM * H_DIM;

    // 8 x 16B per thread per W tile (coalesced 256B runs per 16 lanes)
    auto issueW = [&](int kt, int buf) {
        #pragma unroll
        for (int i = 0; i < 8; ++i) {
            int l   = tid + i * 256;         // 0..2047 float4 slots
            int row = l >> 4;                // 16 float4 per row
            int c4  = (l & 15) << 2;
            copy16_g2l(Wa + (size_t)(r0 + row) * H_DIM + kt + c4,
                       &Ws[buf][row][c4]);
        }
    };
    // 1 x 16B per thread per G tile (list zero-padded => always valid)
    auto issueG = [&](int kt, int buf, int c0) {
        int j = tid >> 4;                    // sample slot 0..15
        int q = (tid & 15) << 2;             // float offset 0..60
        int s = list[c0 + j];
        copy16_g2l(g_prev + (size_t)s * H_DIM + kt + q, &Gs[buf][j][q]);
    };

    for (int c0 = 0; c0 < n; c0 += COLS) {   // sample chunks (usually 1)
        const int nc = min(COLS, n - c0);
        v8f acc = {};

        issueW(0, 0);                        // prologue: tile 0 -> buf 0
        issueG(0, 0, c0);                    // (9 async ops per wave)

        for (int t = 0; t < NT; ++t) {
            const int cur = t & 1;
            if (t + 1 < NT) {                // issue tile t+1 -> other buffer
                issueW((t + 1) * KT, cur ^ 1);
                issueG((t + 1) * KT, cur ^ 1, c0);
                wait_async_le9();            // in-order => tile t landed
            } else {
                wait_async_0();
            }
            __syncthreads();                 // all waves' copies visible

            #pragma unroll
            for (int kk = 0; kk < KT; kk += 4) {
                v2f af, bf;
                af.x = Ws[cur][m0 + lrow][kk + khalf + 0];
                af.y = Ws[cur][m0 + lrow][kk + khalf + 1];
                bf.x = Gs[cur][lrow][kk + khalf + 0];
                bf.y = Gs[cur][lrow][kk + khalf + 1];
                acc = __builtin_amdgcn_wmma_f32_16x16x4_f32(
                        /*neg_a=*/false, af, /*neg_b=*/false, bf,
                        /*c_mod=*/(short)0, acc,
                        /*reuse_a=*/false, /*reuse_b=*/false);
            }
            __syncthreads();                 // done reading buf cur
        }

        // D layout: VGPR v -> M=v (lanes 0-15) / M=v+8 (lanes 16-31), N=lane%16
        if (lrow < nc) {
            int s     = list[c0 + lrow];
            int mbase = r0 + m0 + ((lane >> 4) << 3);
            #pragma unroll
            for (int v = 0; v < 8; ++v) {
                int row = mbase + v;
                out[(size_t)s * H_DIM + row] = tanhf(acc[v] + bias[row]);
            }
        }
    }
}

extern "C" void kernel_launch(void* const* d_in, const int* in_sizes, int n_in,
                              void* d_out, int out_size, void* d_ws, size_t ws_size,
                              hipStream_t stream) {
    const float* g_prev  = (const float*)d_in[0];   // (128, 1024) f32
    const int*   actions = (const int*)  d_in[1];   // (128,) int32
    const float* weights = (const float*)d_in[2];   // (64, 1024, 1024) f32
    const float* bias    = (const float*)d_in[3];   // (1024,) f32
    float*       out     = (float*)d_out;           // (128, 1024) f32
    int*         ws      = (int*)d_ws;              // needs 33,024 bytes

    tm_build_lists<<<1, 256, 0, stream>>>(actions, ws);
    tm_wmma_kernel<<<dim3(A_ACT, H_DIM / ROWS_PB), 256, 0, stream>>>(
        g_prev, weights, bias, ws, out);
}